// SpaceGroupDenseLayer_46428596470438
// MI455X (gfx1250) — compile-verified
//
#include <hip/hip_runtime.h>

typedef __attribute__((ext_vector_type(2))) float v2f;
typedef __attribute__((ext_vector_type(4))) float v4f;
typedef __attribute__((ext_vector_type(8))) float v8f;

#define WAVES 8
#define ROWS_PER_WAVE 16
#define ROWS_PER_BLOCK (WAVES * ROWS_PER_WAVE)   // 128

// LDS weight layout: pair-rows. PAIR_STRIDE pads 64 -> 80 pairs/row so that
// bank(pair r, col) = (32*(r&1) + 2*col) % 64  -> lane halves hit disjoint
// 32-bank sets => conflict-free ds_load_b64 B-fragment reads.
#define PAIR_STRIDE 80
#define SW_FLOATS (32 * PAIR_STRIDE * 2)          // 5120 floats = 20 KB

// Complex [N,32] x [32,32] GEMM realified to [N,64] x [64,64] real GEMM.
//   W_real[2d  ][2f] =  wr   W_real[2d  ][2f+1] = wi
//   W_real[2d+1][2f] = -wi   W_real[2d+1][2f+1] = wr
// Stored in LDS as row pairs: Pair[d][col] = (W_real[2d][col], W_real[2d+1][col])
// so Pair[d][2f] = (wr,-wi), Pair[d][2f+1] = (wi,wr): one b128 store per element.
__global__ __launch_bounds__(256) void sg_dense_wmma(
    const float* __restrict__ in_f,    // inputs  [B,N,32] complex64 -> floats
    const float* __restrict__ w_f,     // weights [230,32,32] complex64 -> floats
    const float* __restrict__ b_f,     // biases  [230,32] complex64 -> floats
    const int*   __restrict__ sg_arr,  // space_group [B]
    float* __restrict__ out_f,         // out [B,N,32] complex64 -> floats
    int Bn, int Nn)
{
    __shared__ float sWp[SW_FLOATS];   // paired + padded realified weight
    __shared__ float sBias[64];

    const int chunks = Nn / ROWS_PER_BLOCK;          // 16
    const int b      = blockIdx.x / chunks;
    const int chunk  = blockIdx.x % chunks;
    const int tid    = threadIdx.x;
    const int sg     = sg_arr[b];

    // ---- Gather + realify weight into LDS (paired layout), one b128 per element ----
    {
        const v2f* wsrc = (const v2f*)w_f + (size_t)sg * 1024;   // 32*32 complex
        #pragma unroll
        for (int i = 0; i < 4; ++i) {
            int e = tid + 256 * i;                  // 0..1023
            int d = e >> 5;
            int f = e & 31;
            v2f wc = wsrc[e];                        // (wr, wi)
            v4f quad = {wc.x, -wc.y, wc.y, wc.x};    // Pair[d][2f], Pair[d][2f+1]
            *(v4f*)&sWp[(d * PAIR_STRIDE + 2 * f) * 2] = quad;   // 16B aligned
        }
        if (tid < 32) {
            v2f bc = ((const v2f*)b_f)[(size_t)sg * 32 + tid];
            sBias[2 * tid]     = bc.x;
            sBias[2 * tid + 1] = bc.y;
        }
    }
    __syncthreads();

    const int wave = tid >> 5;
    const int lane = tid & 31;
    const int m    = lane & 15;   // row-within-tile for A, col-within-tile for B/C
    const int hi   = lane >> 4;   // lane-half selector
    const int n0   = chunk * ROWS_PER_BLOCK + wave * ROWS_PER_WAVE;

    // ---- A fragments for all 16 K-steps (K=4 each over K_total=64) ----
    // ISA 16x4 f32 A layout: VGPR0 = K {0 | 2}, VGPR1 = K {1 | 3} per lane half.
    // Input row n is 64 contiguous floats (interleaved complex) == real GEMM row.
    const float* pA = in_f + ((size_t)b * Nn + n0 + m) * 64 + 2 * hi;
    v2f a[16];
    #pragma unroll
    for (int k = 0; k < 16; ++k)
        a[k] = __builtin_nontemporal_load((const v2f*)(pA + 4 * k)); // streamed once

    // ---- 4 output tiles of 16x16 (cols 0..63 real = 32 complex features) ----
    #pragma unroll
    for (int nt = 0; nt < 4; ++nt) {
        const int col = nt * 16 + m;

        // C init = bias broadcast (bias depends only on col; all 8 C VGPRs share it)
        const float bv = sBias[col];
        v8f c = {bv, bv, bv, bv, bv, bv, bv, bv};

        #pragma unroll
        for (int k = 0; k < 16; ++k) {
            // B fragment rows {4k+2hi, 4k+2hi+1} == Pair[2k+hi][col]: one b64 load.
            v2f bb = *(const v2f*)&sWp[((2 * k + hi) * PAIR_STRIDE + col) * 2];
            c = __builtin_amdgcn_wmma_f32_16x16x4_f32(
                    /*neg_a=*/false, a[k], /*neg_b=*/false, bb,
                    /*c_mod=*/(short)0, c, /*reuse_a=*/false, /*reuse_b=*/false);
        }

        // C layout: VGPR r -> row n0 + r + 8*hi, col = nt*16 + m. Stream out (NT).
        float* po = out_f + ((size_t)b * Nn + n0 + 8 * hi) * 64 + col;
        #pragma unroll
        for (int r = 0; r < 8; ++r)
            __builtin_nontemporal_store(c[r], po + (size_t)r * 64);
    }
}

extern "C" void kernel_launch(void* const* d_in, const int* in_sizes, int n_in,
                              void* d_out, int out_size, void* d_ws, size_t ws_size,
                              hipStream_t stream) {
    (void)in_sizes; (void)n_in; (void)out_size; (void)d_ws; (void)ws_size;
    const float* in_f = (const float*)d_in[0];   // complex64 viewed as float pairs
    const float* w_f  = (const float*)d_in[1];
    const float* b_f  = (const float*)d_in[2];
    const int*   sg   = (const int*)d_in[3];
    float*       out  = (float*)d_out;

    const int B = 512, N = 2048;
    dim3 grid(B * (N / ROWS_PER_BLOCK));   // 8192 blocks
    dim3 block(256);                       // 8 waves of 32
    hipLaunchKernelGGL(sg_dense_wmma, grid, block, 0, stream,
                       in_f, w_f, b_f, sg, out, B, N);
}